// API_RNNAgent_8186207666485
// MI455X (gfx1250) — compile-verified
//
#include <hip/hip_runtime.h>
#include <hip/hip_bf16.h>
#include <math.h>

// ---------------- problem constants ----------------
#define BS    512
#define NA    10
#define NE    11
#define NAL   9
#define H     64
#define HEADS 4
#define EDIM  8
#define ADIM  8
#define OWN   14
#define HYP   64
#define NTOT  (BS*NA)        // 5120
#define NROW_E (NTOT*NE)     // 56320
#define NROW_A (NTOT*NAL)    // 46080
#define KBIG  1040           // 512 (enemy M) + 512 (ally M) + 8 efsum + 8 afsum
#define NBIG  256            // HEADS*H
#define NVP   320            // 260 used (256 vprime + 4 scalar terms), padded to x64
#define NGRU  192

typedef float v2f __attribute__((ext_vector_type(2)));
typedef float v8f __attribute__((ext_vector_type(8)));

// ---------------- workspace layout (float offsets, each 64-aligned) ----------
#define OFF_HEHID   ((size_t)0)                       // 56320*64
#define OFF_HAHID   (OFF_HEHID + (size_t)NROW_E*HYP)  // 46080*64
#define OFF_EMBOWN  (OFF_HAHID + (size_t)NROW_A*HYP)  // 5120*64
#define OFF_M       (OFF_EMBOWN + (size_t)NTOT*H)     // 5120*1040
#define OFF_WBIG    (OFF_M + (size_t)NTOT*KBIG)       // 1040*256
#define OFF_CBIG    (OFF_WBIG + (size_t)KBIG*NBIG)    // 5120*256
#define OFF_WATT    (OFF_CBIG + (size_t)NTOT*NBIG)    // 64*320
#define OFF_BATT    (OFF_WATT + (size_t)H*NVP)        // 320
#define OFF_WIN     (OFF_BATT + (size_t)NVP)          // 256
#define OFF_WOUT    (OFF_WIN + 256)                   // 64 (4 used)
#define OFF_WIHT    (OFF_WOUT + 64)                   // 64*192
#define OFF_WHHT    (OFF_WIHT + (size_t)H*NGRU)       // 64*192
#define OFF_X       (OFF_WHHT + (size_t)H*NGRU)       // 5120*64
#define OFF_GI      (OFF_X + (size_t)NTOT*H)          // 5120*192
#define OFF_GH      (OFF_GI + (size_t)NTOT*NGRU)      // 5120*192
#define OFF_HH      (OFF_GH + (size_t)NTOT*NGRU)      // 5120*64
#define OFF_VP      (OFF_HH + (size_t)NTOT*H)         // 5120*320

#define PREP_TOTAL  (KBIG*NBIG + H*NVP + NVP + H*NGRU + H*NGRU + HEADS*H + HEADS)

// ---------------- weight/packing prep -------------------------------------
__global__ void prep_kernel(const float* __restrict__ he_w2, const float* __restrict__ he_b2,
                            const float* __restrict__ ha_w2, const float* __restrict__ ha_b2,
                            const float* __restrict__ gru_w_ih, const float* __restrict__ gru_w_hh,
                            const float* __restrict__ merge_in_w, const float* __restrict__ merge_out_w,
                            float* __restrict__ Wbig, float* __restrict__ WattE,
                            float* __restrict__ battE, float* __restrict__ WihT,
                            float* __restrict__ WhhT, float* __restrict__ win,
                            float* __restrict__ wout)
{
    int i = blockIdx.x * blockDim.x + threadIdx.x;
    if (i < KBIG * NBIG) {                       // Wbig[1040][256]
        int row = i / NBIG, h = i % NBIG;
        float v;
        if (row < 512)       { int e = row >> 6, k = row & 63; v = he_w2[(e*NBIG + h)*HYP + k]; }
        else if (row < 1024) { int r = row - 512; int e = r >> 6, k = r & 63; v = ha_w2[(e*NBIG + h)*HYP + k]; }
        else if (row < 1032) { v = he_b2[(row - 1024)*NBIG + h]; }
        else                 { v = ha_b2[(row - 1032)*NBIG + h]; }
        Wbig[i] = v;
        return;
    }
    i -= KBIG * NBIG;
    if (i < H * NVP) {                           // WattE[64][320]
        int hrow = i / NVP, c = i % NVP;
        float v = 0.f;
        if (c < 256)      { int head = c >> 6, k = c & 63; v = he_w2[(2048 + head*64 + hrow)*HYP + k]; }
        else if (c < 260) { int head = c - 256;            v = he_b2[2048 + head*64 + hrow]; }
        WattE[i] = v;
        return;
    }
    i -= H * NVP;
    if (i < NVP) {                               // battE[320]
        float v = 0.f;
        if (i < 256)      { int head = i >> 6, k = i & 63; v = he_w2[(2304 + head)*HYP + k]; }
        else if (i < 260) { v = he_b2[2304 + (i - 256)]; }
        battE[i] = v;
        return;
    }
    i -= NVP;
    if (i < H * NGRU) { int k = i / NGRU, c = i % NGRU; WihT[i] = gru_w_ih[c*H + k]; return; }
    i -= H * NGRU;
    if (i < H * NGRU) { int k = i / NGRU, c = i % NGRU; WhhT[i] = gru_w_hh[c*H + k]; return; }
    i -= H * NGRU;
    if (i < HEADS * H) {                         // softmax over heads of merge_in_w
        int h = i & 63;
        float m0 = merge_in_w[0*H+h], m1 = merge_in_w[1*H+h];
        float m2 = merge_in_w[2*H+h], m3 = merge_in_w[3*H+h];
        float mx = fmaxf(fmaxf(m0, m1), fmaxf(m2, m3));
        float e0 = expf(m0-mx), e1 = expf(m1-mx), e2 = expf(m2-mx), e3 = expf(m3-mx);
        float inv = 1.f / (e0+e1+e2+e3);
        float ev = (i>>6)==0 ? e0 : (i>>6)==1 ? e1 : (i>>6)==2 ? e2 : e3;
        win[i] = ev * inv;
        return;
    }
    i -= HEADS * H;
    if (i < HEADS) {                             // softmax of merge_out_w
        float m0 = merge_out_w[0], m1 = merge_out_w[1], m2 = merge_out_w[2], m3 = merge_out_w[3];
        float mx = fmaxf(fmaxf(m0, m1), fmaxf(m2, m3));
        float e0 = expf(m0-mx), e1 = expf(m1-mx), e2 = expf(m2-mx), e3 = expf(m3-mx);
        float inv = 1.f / (e0+e1+e2+e3);
        float ev = i==0 ? e0 : i==1 ? e1 : i==2 ? e2 : e3;
        wout[i] = ev * inv;
    }
}

// ---------------- hypernet hidden: relu(feats @ w1^T + b1), K=8 -------------
__global__ __launch_bounds__(256)
void hyper_hidden_kernel(const float* __restrict__ feats, const float* __restrict__ w1,
                         const float* __restrict__ b1, float* __restrict__ out, int nrows)
{
    __shared__ float sw[HYP*EDIM];
    __shared__ float sb[HYP];
    for (int i = threadIdx.x; i < HYP*EDIM; i += blockDim.x) sw[i] = w1[i];
    if (threadIdx.x < HYP) sb[threadIdx.x] = b1[threadIdx.x];
    __syncthreads();
    int row = blockIdx.x * blockDim.x + threadIdx.x;
    if (row >= nrows) return;
    float f[EDIM];
#pragma unroll
    for (int e = 0; e < EDIM; ++e) f[e] = feats[(size_t)row*EDIM + e];
    float* o = out + (size_t)row * HYP;
#pragma unroll 4
    for (int c = 0; c < HYP; ++c) {
        float acc = sb[c];
#pragma unroll
        for (int e = 0; e < EDIM; ++e) acc = fmaf(f[e], sw[c*EDIM + e], acc);
        o[c] = fmaxf(acc, 0.f);
    }
}

// ---------------- own embedding + id/action embedding gathers ---------------
__global__ void emb_own_kernel(const float* __restrict__ own, const float* __restrict__ w,
                               const float* __restrict__ b, const float* __restrict__ agent_emb,
                               const float* __restrict__ action_emb, const int* __restrict__ aidx,
                               const int* __restrict__ lidx, float* __restrict__ embown)
{
    int i = blockIdx.x * blockDim.x + threadIdx.x;
    if (i >= NTOT * H) return;
    int n = i >> 6, h = i & 63;
    float acc = b[h];
    const float* of = own + (size_t)n * OWN;
    const float* wr = w + h * OWN;
#pragma unroll
    for (int o = 0; o < OWN; ++o) acc = fmaf(of[o], wr[o], acc);
    acc += agent_emb[aidx[n]*H + h] + action_emb[lidx[n]*H + h];
    embown[i] = acc;
}

// -------- per-agent outer products M[n] = [ef^T@hehid | af^T@hahid | sums] --
__global__ __launch_bounds__(64)
void mbuild_kernel(const float* __restrict__ ef, const float* __restrict__ af,
                   const float* __restrict__ hehid, const float* __restrict__ hahid,
                   float* __restrict__ M)
{
    __shared__ float sef[NE*EDIM];   // 88
    __shared__ float saf[NAL*ADIM];  // 72
    int n = blockIdx.x, t = threadIdx.x;
    for (int i = t; i < NE*EDIM;  i += 64) sef[i] = ef[(size_t)n*NE*EDIM  + i];
    for (int i = t; i < NAL*ADIM; i += 64) saf[i] = af[(size_t)n*NAL*ADIM + i];
    __syncthreads();
    float* Mrow = M + (size_t)n * KBIG;
    float hv[NE];
#pragma unroll
    for (int j = 0; j < NE; ++j) hv[j] = hehid[((size_t)(n*NE + j))*HYP + t];
#pragma unroll
    for (int e = 0; e < EDIM; ++e) {
        float acc = 0.f;
#pragma unroll
        for (int j = 0; j < NE; ++j) acc = fmaf(sef[j*EDIM + e], hv[j], acc);
        Mrow[e*HYP + t] = acc;
    }
    float av[NAL];
#pragma unroll
    for (int j = 0; j < NAL; ++j) av[j] = hahid[((size_t)(n*NAL + j))*HYP + t];
#pragma unroll
    for (int e = 0; e < ADIM; ++e) {
        float acc = 0.f;
#pragma unroll
        for (int j = 0; j < NAL; ++j) acc = fmaf(saf[j*ADIM + e], av[j], acc);
        Mrow[512 + e*HYP + t] = acc;
    }
    if (t < EDIM) {
        float s = 0.f;
#pragma unroll
        for (int j = 0; j < NE; ++j) s += sef[j*EDIM + t];
        Mrow[1024 + t] = s;
        float s2 = 0.f;
#pragma unroll
        for (int j = 0; j < NAL; ++j) s2 += saf[j*ADIM + t];
        Mrow[1032 + t] = s2;
    }
}

// ---------------- generic FP32 WMMA GEMM: C[M,N] = A[M,K] @ B[K,N] (+bias) --
// wave tile 16x64 via V_WMMA_F32_16X16X4_F32; block = 4 waves = 64x64 tile.
// Requires M%64==0, N%64==0, K%4==0; no divergence -> EXEC all-ones.
__global__ __launch_bounds__(128)
void wmma_gemm_f32_kernel(const float* __restrict__ A, const float* __restrict__ B,
                          const float* __restrict__ bias, float* __restrict__ C,
                          int K, int lda, int ldb, int ldc)
{
    const int lane  = threadIdx.x & 31;
    const int wave  = threadIdx.x >> 5;
    const int mbase = blockIdx.x * 64 + wave * 16;
    const int nbase = blockIdx.y * 64;
    const int l15   = lane & 15;
    const int koff  = (lane >> 4) << 1;        // 0 (lanes 0-15) / 2 (lanes 16-31)

    const float* Arow = A + (size_t)(mbase + l15) * lda;
    v8f acc0 = {}, acc1 = {}, acc2 = {}, acc3 = {};

    for (int k = 0; k < K; k += 4) {
        v2f av = *(const v2f*)(Arow + k + koff);                 // A[m, k+koff .. +1]
        const float* b0p = B + (size_t)(k + koff) * ldb + nbase + l15;
        const float* b1p = b0p + ldb;
        v2f b0; b0.x = b0p[0];  b0.y = b1p[0];
        v2f b1; b1.x = b0p[16]; b1.y = b1p[16];
        v2f b2; b2.x = b0p[32]; b2.y = b1p[32];
        v2f b3; b3.x = b0p[48]; b3.y = b1p[48];
        acc0 = __builtin_amdgcn_wmma_f32_16x16x4_f32(false, av, false, b0, (short)0, acc0, false, false);
        acc1 = __builtin_amdgcn_wmma_f32_16x16x4_f32(false, av, false, b1, (short)0, acc1, false, false);
        acc2 = __builtin_amdgcn_wmma_f32_16x16x4_f32(false, av, false, b2, (short)0, acc2, false, false);
        acc3 = __builtin_amdgcn_wmma_f32_16x16x4_f32(false, av, false, b3, (short)0, acc3, false, false);
    }

    float bb0 = 0.f, bb1 = 0.f, bb2 = 0.f, bb3 = 0.f;
    if (bias) {
        bb0 = bias[nbase + l15];      bb1 = bias[nbase + l15 + 16];
        bb2 = bias[nbase + l15 + 32]; bb3 = bias[nbase + l15 + 48];
    }
    const int rbase = mbase + ((lane >> 4) << 3);   // +8 rows for lanes 16-31
#pragma unroll
    for (int v = 0; v < 8; ++v) {
        float* Crow = C + (size_t)(rbase + v) * ldc + nbase + l15;
        Crow[0]  = acc0[v] + bb0;
        Crow[16] = acc1[v] + bb1;
        Crow[32] = acc2[v] + bb2;
        Crow[48] = acc3[v] + bb3;
    }
}

// ---------------- softmax-head merge + relu -> GRU input x ------------------
__global__ void merge_x_kernel(const float* __restrict__ embown, const float* __restrict__ Cbig,
                               const float* __restrict__ win, float* __restrict__ x)
{
    int i = blockIdx.x * blockDim.x + threadIdx.x;
    if (i >= NTOT * H) return;
    int n = i >> 6, h = i & 63;
    const float* crow = Cbig + (size_t)n * NBIG;
    float acc = embown[i];
#pragma unroll
    for (int head = 0; head < HEADS; ++head)
        acc = fmaf(win[head*H + h], crow[head*H + h], acc);
    x[i] = fmaxf(acc, 0.f);
}

// ---------------- GRU gate nonlinearity (torch order: r,z,n) ---------------
__global__ void gru_finish_kernel(const float* __restrict__ gi, const float* __restrict__ gh,
                                  const float* __restrict__ hin, float* __restrict__ hh)
{
    int i = blockIdx.x * blockDim.x + threadIdx.x;
    if (i >= NTOT * H) return;
    int n = i >> 6, h = i & 63;
    const float* gir = gi + (size_t)n * NGRU;
    const float* ghr = gh + (size_t)n * NGRU;
    float r  = 1.f / (1.f + expf(-(gir[h]       + ghr[h])));
    float z  = 1.f / (1.f + expf(-(gir[H + h]   + ghr[H + h])));
    float nn = tanhf(gir[2*H + h] + r * ghr[2*H + h]);
    hh[i] = (1.f - z) * nn + z * hin[i];
}

// ---------------- q_normal + merged attack Q -------------------------------
__global__ __launch_bounds__(64)
void final_kernel(const float* __restrict__ hh, const float* __restrict__ vprime,
                  const float* __restrict__ hehid, const float* __restrict__ fc2_w,
                  const float* __restrict__ fc2_b, const float* __restrict__ wout,
                  float* __restrict__ out)
{
    __shared__ float shh[H];
    __shared__ float svp[260];
    __shared__ float swo[HEADS];
    int n = blockIdx.x, t = threadIdx.x;
    shh[t] = hh[(size_t)n*H + t];
    for (int i = t; i < 260; i += 64) svp[i] = vprime[(size_t)n*NVP + i];
    if (t < HEADS) swo[t] = wout[t];
    __syncthreads();
    if (t < 6) {                                   // q_normal
        float acc = fc2_b[t];
        const float* wr = fc2_w + t*H;
        for (int k = 0; k < H; ++k) acc = fmaf(shh[k], wr[k], acc);
        out[(size_t)n*17 + t] = acc;
    } else if (t >= 16 && t < 16 + NE) {           // q_attack, one enemy each
        int j = t - 16;
        const float* eh = hehid + ((size_t)(n*NE + j))*HYP;
        float q = 0.f;
#pragma unroll
        for (int head = 0; head < HEADS; ++head) {
            float d = svp[256 + head];             // per-(n,head) scalar term
            const float* vp = svp + head*H;
            for (int k = 0; k < H; ++k) d = fmaf(eh[k], vp[k], d);
            q = fmaf(swo[head], d, q);
        }
        out[(size_t)n*17 + 6 + j] = q;
    }
}

// ---------------- launcher --------------------------------------------------
extern "C" void kernel_launch(void* const* d_in, const int* in_sizes, int n_in,
                              void* d_out, int out_size, void* d_ws, size_t ws_size,
                              hipStream_t stream)
{
    (void)in_sizes; (void)n_in; (void)out_size; (void)ws_size;
    const float* own_feats   = (const float*)d_in[0];
    const float* enemy_feats = (const float*)d_in[1];
    const float* ally_feats  = (const float*)d_in[2];
    const float* hidden      = (const float*)d_in[3];
    const int*   aidx        = (const int*)d_in[4];
    const int*   lidx        = (const int*)d_in[5];
    const float* fc1_own_w   = (const float*)d_in[6];
    const float* fc1_own_b   = (const float*)d_in[7];
    const float* agent_emb   = (const float*)d_in[8];
    const float* action_emb  = (const float*)d_in[9];
    const float* he_w1       = (const float*)d_in[10];
    const float* he_b1       = (const float*)d_in[11];
    const float* he_w2       = (const float*)d_in[12];
    const float* he_b2       = (const float*)d_in[13];
    const float* ha_w1       = (const float*)d_in[14];
    const float* ha_b1       = (const float*)d_in[15];
    const float* ha_w2       = (const float*)d_in[16];
    const float* ha_b2       = (const float*)d_in[17];
    const float* merge_in_w  = (const float*)d_in[18];
    const float* gru_w_ih    = (const float*)d_in[19];
    const float* gru_w_hh    = (const float*)d_in[20];
    const float* gru_b_ih    = (const float*)d_in[21];
    const float* gru_b_hh    = (const float*)d_in[22];
    const float* fc2_w       = (const float*)d_in[23];
    const float* fc2_b       = (const float*)d_in[24];
    const float* merge_out_w = (const float*)d_in[25];

    float* ws     = (float*)d_ws;
    float* hehid  = ws + OFF_HEHID;
    float* hahid  = ws + OFF_HAHID;
    float* embown = ws + OFF_EMBOWN;
    float* Mmat   = ws + OFF_M;
    float* Wbig   = ws + OFF_WBIG;
    float* Cbig   = ws + OFF_CBIG;
    float* WattE  = ws + OFF_WATT;
    float* battE  = ws + OFF_BATT;
    float* win    = ws + OFF_WIN;
    float* wout   = ws + OFF_WOUT;
    float* WihT   = ws + OFF_WIHT;
    float* WhhT   = ws + OFF_WHHT;
    float* xb     = ws + OFF_X;
    float* gib    = ws + OFF_GI;
    float* ghb    = ws + OFF_GH;
    float* hhb    = ws + OFF_HH;
    float* vp     = ws + OFF_VP;

    // 1) pack/transpose weights, softmax merge weights
    prep_kernel<<<(PREP_TOTAL + 255)/256, 256, 0, stream>>>(
        he_w2, he_b2, ha_w2, ha_b2, gru_w_ih, gru_w_hh, merge_in_w, merge_out_w,
        Wbig, WattE, battE, WihT, WhhT, win, wout);

    // 2) hypernet hidden layers (relu), K=8
    hyper_hidden_kernel<<<(NROW_E + 255)/256, 256, 0, stream>>>(enemy_feats, he_w1, he_b1, hehid, NROW_E);
    hyper_hidden_kernel<<<(NROW_A + 255)/256, 256, 0, stream>>>(ally_feats,  ha_w1, ha_b1, hahid, NROW_A);

    // 3) own embedding + gathers
    emb_own_kernel<<<(NTOT*H + 255)/256, 256, 0, stream>>>(
        own_feats, fc1_own_w, fc1_own_b, agent_emb, action_emb, aidx, lidx, embown);

    // 4) per-agent feature x hidden outer products (+ feature sums for bias rows)
    mbuild_kernel<<<NTOT, 64, 0, stream>>>(enemy_feats, ally_feats, hehid, hahid, Mmat);

    // 5) big fused entity GEMM: [5120 x 1040] @ [1040 x 256] -> emb_e+emb_a per head
    dim3 gBig(NTOT/64, NBIG/64);
    wmma_gemm_f32_kernel<<<gBig, 128, 0, stream>>>(Mmat, Wbig, nullptr, Cbig, KBIG, KBIG, NBIG, NBIG);

    // 6) gh = hidden @ W_hh^T + b_hh  (independent of x)
    dim3 gGru(NTOT/64, NGRU/64);
    wmma_gemm_f32_kernel<<<gGru, 128, 0, stream>>>(hidden, WhhT, gru_b_hh, ghb, H, H, NGRU, NGRU);

    // 7) softmax-head merge + relu -> x
    merge_x_kernel<<<(NTOT*H + 255)/256, 256, 0, stream>>>(embown, Cbig, win, xb);

    // 8) gi = x @ W_ih^T + b_ih
    wmma_gemm_f32_kernel<<<gGru, 128, 0, stream>>>(xb, WihT, gru_b_ih, gib, H, H, NGRU, NGRU);

    // 9) GRU gates -> hh
    gru_finish_kernel<<<(NTOT*H + 255)/256, 256, 0, stream>>>(gib, ghb, hidden, hhb);

    // 10) v' = hh @ W_att [64 x 320] + b_att (cols 256..259 carry the scalar terms)
    dim3 gVp(NTOT/64, NVP/64);
    wmma_gemm_f32_kernel<<<gVp, 128, 0, stream>>>(hhb, WattE, battE, vp, H, H, NVP, NVP);

    // 11) q_normal + q_attack -> out [512,10,17]
    final_kernel<<<NTOT, 64, 0, stream>>>(hhb, vp, hehid, fc2_w, fc2_b, wout, (float*)d_out);
}